// CrfLogLikelihood_30915174596796
// MI455X (gfx1250) — compile-verified
//
#include <hip/hip_runtime.h>
#include <hip/hip_bf16.h>
#include <math.h>

#define BB 512
#define TT 512
#define NN 64
#define STRIDE 68   // padded LDS row stride (floats): 16B-aligned rows, conflict-free WMMA A loads

typedef __attribute__((ext_vector_type(2))) float v2f;
typedef __attribute__((ext_vector_type(8))) float v8f;

// Monotone float<->uint key so unsigned atomicMax implements float max.
__device__ __forceinline__ unsigned fkey(float x) {
    unsigned b = __float_as_uint(x);
    return (b & 0x80000000u) ? ~b : (b | 0x80000000u);
}
__device__ __forceinline__ float funkey(unsigned k) {
    unsigned b = (k & 0x80000000u) ? (k ^ 0x80000000u) : ~k;
    return __uint_as_float(b);
}

// ---------------------------------------------------------------------------
// Kernel 1: CRF sequence score (gather + masked sum). One wave per batch.
// ---------------------------------------------------------------------------
__global__ void seq_score_kernel(const float* __restrict__ inputs,
                                 const float* __restrict__ trans,
                                 const int*   __restrict__ tags,
                                 const int*   __restrict__ lens,
                                 float*       __restrict__ ws) {
    int b    = blockIdx.x;
    int lane = threadIdx.x;          // 0..31
    int len  = lens[b];
    const int*   tg = tags + b * TT;
    const float* in = inputs + (size_t)b * TT * NN;
    float acc = 0.0f;
    for (int t = lane; t < TT; t += 32) {
        if (t < len) {
            int tag = tg[t];
            acc += in[t * NN + tag];                 // unary, mask t < len
            if (t >= 1) acc += trans[tg[t - 1] * NN + tag];  // binary, mask[:,1:]
        }
    }
    for (int off = 16; off > 0; off >>= 1) acc += __shfl_down(acc, off, 32);
    if (lane == 0) ws[b] = acc;
}

// ---------------------------------------------------------------------------
// Kernel 2: CRF forward recursion with V_WMMA_F32_16X16X4_F32.
// Block = 16 batches, 128 threads (4 waves). Wave w computes output N-tile w.
// ---------------------------------------------------------------------------
__global__ void __launch_bounds__(128)
crf_forward_wmma(const float* __restrict__ inputs,
                 const float* __restrict__ trans,
                 const int*   __restrict__ lens,
                 const float* __restrict__ ws,
                 float*       __restrict__ out) {
    __shared__ float    A[16 * STRIDE];   // alpha[row][col]
    __shared__ float    P[16 * STRIDE];   // exp(alpha - m)
    __shared__ float    F[16 * STRIDE];   // alpha captured at last_index
    __shared__ unsigned MK[2][16];        // double-buffered row-max keys

    const int tid  = threadIdx.x;
    const int wave = tid >> 5;            // 0..3 -> output N-tile
    const int lane = tid & 31;
    const int h    = lane >> 4;           // half-wave
    const int l16  = lane & 15;
    const int col  = wave * 16 + l16;     // this thread's output column
    const int b0   = blockIdx.x * 16;

    // Preload B fragments: E = exp(trans), WMMA B layout for 16x16x4:
    // lane L, VGPR v holds E[4*kt + v + 2*(L/16)][col]. Loop-invariant -> VGPRs.
    v2f Bf[16];
#pragma unroll
    for (int kt = 0; kt < 16; ++kt) {
        int r0   = 4 * kt + 2 * h;
        Bf[kt].x = __expf(trans[r0 * NN + col]);
        Bf[kt].y = __expf(trans[(r0 + 1) * NN + col]);
    }

    // Per-thread row metadata (D layout: reg r <-> row r + 8*h).
    int          li[8];
    const float* ebase[8];
#pragma unroll
    for (int r = 0; r < 8; ++r) {
        int row  = r + 8 * h;
        int L    = lens[b0 + row];
        li[r]    = (L > 1) ? (L - 1) : 0;                       // max(0, len-1)
        ebase[r] = inputs + (size_t)(b0 + row) * TT * NN + col; // emit pointer
    }

    // t = 0: alpha = first emissions; F covers last_index == 0.
#pragma unroll
    for (int r = 0; r < 8; ++r) {
        int   row = r + 8 * h;
        float v   = ebase[r][0];
        A[row * STRIDE + col] = v;
        F[row * STRIDE + col] = v;
    }
    if (tid < 32) MK[tid >> 4][tid & 15] = 0u;
    __syncthreads();

    for (int t = 1; t < TT; ++t) {
        const int cur = t & 1;

        // ---- Phase A: row max (keyed atomicMax), then P = exp(alpha - m) ----
        const int    arow = tid & 15;
        const int    aseg = tid >> 4;                  // 8 lanes x 8 cols per row
        const float* arp  = &A[arow * STRIDE + aseg * 8];
        float mx = arp[0];
#pragma unroll
        for (int i = 1; i < 8; ++i) mx = fmaxf(mx, arp[i]);
        atomicMax(&MK[cur][arow], fkey(mx));
        if (tid < 16) MK[cur ^ 1][tid] = 0u;           // prep next buffer
        __syncthreads();

        const float m  = funkey(MK[cur][arow]);
        float*      pr = &P[arow * STRIDE + aseg * 8];
#pragma unroll
        for (int i = 0; i < 8; ++i) pr[i] = __expf(arp[i] - m);
        __syncthreads();

        // ---- Phase B: S(16x16 tile) = P(16x64) @ E(64x64), K chained by 4 ----
        v8f c = {0.f, 0.f, 0.f, 0.f, 0.f, 0.f, 0.f, 0.f};
#pragma unroll
        for (int kt = 0; kt < 16; ++kt) {
            // A frag: lane L holds P[l16][4*kt + v + 2*h], v = 0,1 (8B aligned)
            v2f a = *(const v2f*)&P[l16 * STRIDE + 4 * kt + 2 * h];
            c = __builtin_amdgcn_wmma_f32_16x16x4_f32(
                    false, a, false, Bf[kt], (short)0, c, false, false);
        }

        // ---- Phase C: alpha' = emit + m + log(S), write back in D layout ----
#pragma unroll
        for (int r = 0; r < 8; ++r) {
            int   row  = r + 8 * h;
            float mrow = funkey(MK[cur][row]);
            float emit = ebase[r][(size_t)t * NN];
            float nv   = emit + mrow + __logf(c[r]);
            A[row * STRIDE + col] = nv;
            if (li[r] == t) F[row * STRIDE + col] = nv;
        }
        __syncthreads();
    }

    // Epilogue: log_norm = logsumexp(F[row]); out = seq_score - log_norm.
    if (tid < 16) {
        int   row = tid;
        float mx  = F[row * STRIDE];
        for (int n = 1; n < NN; ++n) mx = fmaxf(mx, F[row * STRIDE + n]);
        float s = 0.0f;
        for (int n = 0; n < NN; ++n) s += __expf(F[row * STRIDE + n] - mx);
        float logz = mx + __logf(s);
        out[b0 + row] = ws[b0 + row] - logz;
    }
}

// ---------------------------------------------------------------------------
extern "C" void kernel_launch(void* const* d_in, const int* in_sizes, int n_in,
                              void* d_out, int out_size, void* d_ws, size_t ws_size,
                              hipStream_t stream) {
    const float* inputs = (const float*)d_in[0];   // (512,512,64) f32
    const float* trans  = (const float*)d_in[1];   // (64,64) f32
    const int*   tags   = (const int*)d_in[2];     // (512,512) i32
    const int*   lens   = (const int*)d_in[3];     // (512,) i32
    float* out = (float*)d_out;                    // [0,512): ll, [512,4608): trans
    float* ws  = (float*)d_ws;                     // [0,512): sequence scores

    hipMemcpyAsync(out + BB, trans, NN * NN * sizeof(float),
                   hipMemcpyDeviceToDevice, stream);
    seq_score_kernel<<<BB, 32, 0, stream>>>(inputs, trans, tags, lens, ws);
    crf_forward_wmma<<<BB / 16, 128, 0, stream>>>(inputs, trans, lens, ws, out);
}